// Cross_modal_Center_ContrastiveLoss_36618891166025
// MI455X (gfx1250) — compile-verified
//
#include <hip/hip_runtime.h>
#include <stdint.h>

#ifndef __has_builtin
#define __has_builtin(x) 0
#endif

typedef int v4i __attribute__((vector_size(16)));
typedef __attribute__((address_space(1))) v4i gv4i;
typedef __attribute__((address_space(3))) v4i lv4i;

// ---------------- async global -> LDS path selection ----------------
#define TRY_ASM_ASYNC 1
#if __has_builtin(__builtin_amdgcn_global_load_async_to_lds_b128)
#  define ASYNC_MODE 2   // clang builtin (confirmed present on this toolchain)
#elif defined(TRY_ASM_ASYNC) && defined(__gfx1250__)
#  define ASYNC_MODE 1   // inline asm fallback
#else
#  define ASYNC_MODE 0   // plain loads
#endif

#if ASYNC_MODE
__device__ __forceinline__ void async_wait(int n) {
#if __has_builtin(__builtin_amdgcn_s_wait_asynccnt)
  if (n == 0) __builtin_amdgcn_s_wait_asynccnt(0);
  else        __builtin_amdgcn_s_wait_asynccnt(2);
#else
  if (n == 0) asm volatile("s_wait_asynccnt 0" ::: "memory");
  else        asm volatile("s_wait_asynccnt 2" ::: "memory");
#endif
}

__device__ __forceinline__ void ds_wait0() {
#if __has_builtin(__builtin_amdgcn_s_wait_dscnt)
  __builtin_amdgcn_s_wait_dscnt(0);
#else
  asm volatile("s_wait_dscnt 0" ::: "memory");
#endif
}

__device__ __forceinline__ void async_copy16(const float* g, float* l) {
#if ASYNC_MODE == 2
  __builtin_amdgcn_global_load_async_to_lds_b128((gv4i*)g, (lv4i*)l, 0, 0);
#else
  asm volatile("global_load_async_to_lds_b128 %0, %1, off"
               :: "v"((uint32_t)(uintptr_t)l), "v"(g) : "memory");
#endif
}
#endif // ASYNC_MODE

#define C_PAD 512
#define ROWS_PER_STAGE 4

// ---------------- main: per-class segment mean + weighted Huber ----------------
// grid = (C, D/128), block = 128 (4 independent waves; no block barriers needed).
// Wave w owns features [blockIdx.y*128 + w*32, +32) of one class and streams the
// class's rows 4 at a time: one global_load_async_to_lds_b128 (32 lanes x 16B)
// fetches 4 rows x 128B per modality, double-buffered on s_wait_asynccnt.
__global__ void k_main(const float* __restrict__ m1, const float* __restrict__ m2,
                       const float* __restrict__ centers,
                       const int* __restrict__ counts, const int* __restrict__ offsets,
                       const int* __restrict__ idx, int D, float* loss_acc) {
  const int c    = blockIdx.x;
  const int wave = threadIdx.x >> 5;
  const int lane = threadIdx.x & 31;
  const int fb   = blockIdx.y * 128 + wave * 32;   // wave's feature base
  const int f    = fb + lane;                      // this lane's feature
  const int cnt  = counts[c];

  float s1 = 0.f, s2 = 0.f;
  if (cnt > 0) {
    const int start = offsets[c];
#if ASYNC_MODE
    __shared__ float lds[4][2][2][ROWS_PER_STAGE * 32];  // [wave][buf][modality][row*32+feat]
    const int rsub = lane >> 3;      // which of the 4 rows this lane fetches for
    const int q4   = (lane & 7) * 4; // float offset of this lane's 16B chunk in the 128B row slice
    const int full = cnt & ~(ROWS_PER_STAGE - 1);

    if (full > 0) {
      int buf = 0;
      {
        int gi = idx[start + rsub];
        async_copy16(m1 + (size_t)gi * D + fb + q4, &lds[wave][0][0][rsub * 32 + q4]);
        async_copy16(m2 + (size_t)gi * D + fb + q4, &lds[wave][0][1][rsub * 32 + q4]);
      }
      for (int r = ROWS_PER_STAGE; r < full; r += ROWS_PER_STAGE) {
        int gi = idx[start + r + rsub];
        int nb = buf ^ 1;
        // Buffer `nb` was consumed 1 iteration ago via ds_loads; make sure those
        // DS reads retired before async writes can land in the same LDS lines.
        ds_wait0();
        async_copy16(m1 + (size_t)gi * D + fb + q4, &lds[wave][nb][0][rsub * 32 + q4]);
        async_copy16(m2 + (size_t)gi * D + fb + q4, &lds[wave][nb][1][rsub * 32 + q4]);
        async_wait(2);  // the 2 newest may remain in flight; current buffer is ready
        #pragma unroll
        for (int rr = 0; rr < ROWS_PER_STAGE; ++rr) {
          s1 += lds[wave][buf][0][rr * 32 + lane];
          s2 += lds[wave][buf][1][rr * 32 + lane];
        }
        buf = nb;
      }
      async_wait(0);
      #pragma unroll
      for (int rr = 0; rr < ROWS_PER_STAGE; ++rr) {
        s1 += lds[wave][buf][0][rr * 32 + lane];
        s2 += lds[wave][buf][1][rr * 32 + lane];
      }
    }
    for (int rr = full; rr < cnt; ++rr) {  // remainder rows (0..3)
      int gi = idx[start + rr];
      s1 += m1[(size_t)gi * D + f];
      s2 += m2[(size_t)gi * D + f];
    }
#else
    for (int rr = 0; rr < cnt; ++rr) {
      int gi = idx[start + rr];
      s1 += m1[(size_t)gi * D + f];
      s2 += m2[(size_t)gi * D + f];
    }
#endif
  }

  float v = 0.f;
  if (cnt > 0) {
    float fc  = (float)cnt;
    float ctr = centers[(size_t)c * D + f];
    float d1 = fabsf(s1 / fc - ctr);
    float d2 = fabsf(s2 / fc - ctr);
    float h1 = d1 < 1.f ? 0.5f * d1 * d1 : d1 - 0.5f;
    float h2 = d2 < 1.f ? 0.5f * d2 * d2 : d2 - 0.5f;
    v = (h1 + h2) * fc;  // this class's Huber sum is weighted by count[c]
  }
  // wave32 reduction
  for (int o = 16; o > 0; o >>= 1) v += __shfl_down(v, o, 32);
  if (lane == 0 && cnt > 0) atomicAdd(loss_acc, v);
}

// ---------------- tiny pipeline kernels ----------------
__global__ void k_init(int* counts, float* loss_acc) {
  counts[threadIdx.x] = 0;
  if (threadIdx.x == 0) *loss_acc = 0.f;
}

__global__ void k_hist(const int* __restrict__ targets, int n, int* counts) {
  int i = blockIdx.x * blockDim.x + threadIdx.x;
  if (i < n) atomicAdd(&counts[targets[i]], 1);
}

__global__ void k_scan(const int* __restrict__ counts, int* offsets, int* cursor) {
  __shared__ int tmp[C_PAD];
  int t = threadIdx.x;
  tmp[t] = counts[t];
  __syncthreads();
  for (int off = 1; off < C_PAD; off <<= 1) {
    int v = (t >= off) ? tmp[t - off] : 0;
    __syncthreads();
    tmp[t] += v;
    __syncthreads();
  }
  int excl = (t == 0) ? 0 : tmp[t - 1];
  offsets[t] = excl;
  cursor[t]  = excl;
}

__global__ void k_scatter(const int* __restrict__ targets, int n, int* cursor, int* idx) {
  int i = blockIdx.x * blockDim.x + threadIdx.x;
  if (i < n) {
    int pos = atomicAdd(&cursor[targets[i]], 1);
    idx[pos] = i;
  }
}

__global__ void k_final(const float* loss_acc, float* out, float inv) {
  out[0] = loss_acc[0] * inv;
}

// ---------------- host launcher ----------------
extern "C" void kernel_launch(void* const* d_in, const int* in_sizes, int n_in,
                              void* d_out, int out_size, void* d_ws, size_t ws_size,
                              hipStream_t stream) {
  const float* m1      = (const float*)d_in[0];
  const float* m2      = (const float*)d_in[1];
  const float* centers = (const float*)d_in[2];
  const int*   targets = (const int*)d_in[3];

  const int N = in_sizes[3];
  const int D = in_sizes[0] / N;       // 512
  const int C = in_sizes[2] / D;       // 395

  char* ws = (char*)d_ws;
  int*   counts   = (int*)(ws + 0);
  int*   offsets  = (int*)(ws + 2048);
  int*   cursor   = (int*)(ws + 4096);
  float* loss_acc = (float*)(ws + 6144);
  int*   idx      = (int*)(ws + 6400);

  k_init   <<<1, C_PAD, 0, stream>>>(counts, loss_acc);
  k_hist   <<<(N + 255) / 256, 256, 0, stream>>>(targets, N, counts);
  k_scan   <<<1, C_PAD, 0, stream>>>(counts, offsets, cursor);
  k_scatter<<<(N + 255) / 256, 256, 0, stream>>>(targets, N, cursor, idx);

  dim3 grid(C, D / 128);
  k_main   <<<grid, 128, 0, stream>>>(m1, m2, centers, counts, offsets, idx, D, loss_acc);
  k_final  <<<1, 1, 0, stream>>>(loss_acc, (float*)d_out, 1.0f / ((float)N * (float)D));
}